// MatrixNetwork_50173807952657
// MI455X (gfx1250) — compile-verified
//
#include <hip/hip_runtime.h>

#define Nn    64
#define VOCAB 12
#define SEQ   16384
#define NBLK  128
#define CHUNK (SEQ / NBLK)   // 128 matrices per block

typedef float v2f __attribute__((ext_vector_type(2)));
typedef float v8f __attribute__((ext_vector_type(8)));

// ---------------------------------------------------------------------------
// Phase 1: each block computes the ordered product of CHUNK gathered 64x64
// matrices using fp32 WMMA (V_WMMA_F32_16X16X4_F32), keeping the running
// product P in LDS. 16 waves, one 16x16 output tile each. The next matrix is
// prefetched into registers while the WMMA chain of the current step runs,
// and committed to a double-buffered LDS staging area after the read fence.
// ---------------------------------------------------------------------------
__global__ __launch_bounds__(512) void chain_partial_kernel(
    const float* __restrict__ token_mats,   // [12, 64, 64] row-major
    const int*   __restrict__ token_ids,    // [SEQ]
    float*       __restrict__ partials)     // [NBLK, 64, 64]
{
    __shared__ float P [Nn * Nn];        // running product       (16 KB)
    __shared__ float Mb[2][Nn * Nn];     // double-buffered M_t   (32 KB)
    __shared__ int   ids[CHUNK];         // chunk token ids

    const int tid  = threadIdx.x;
    const int lane = tid & 31;
    const int wave = tid >> 5;       // 0..15
    const int ti   = wave >> 2;      // output row-tile  (0..3)
    const int tj   = wave & 3;       // output col-tile  (0..3)
    const int half = lane >> 4;      // lane group 0/1
    const int n16  = lane & 15;

    const int t0 = blockIdx.x * CHUNK;

    // stage token ids + P = Identity
    if (tid < CHUNK) ids[tid] = token_ids[t0 + tid];
    for (int idx = tid; idx < Nn * Nn; idx += 512)
        P[idx] = ((idx >> 6) == (idx & 63)) ? 1.0f : 0.0f;
    __syncthreads();

    // preload first matrix into buffer 0 (visible after first loop barrier)
    {
        const float4* src = (const float4*)(token_mats + (size_t)ids[0] * (Nn * Nn));
        float4* dst = (float4*)Mb[0];
        dst[tid]       = src[tid];
        dst[tid + 512] = src[tid + 512];
    }

    const int arow = (ti * 16 + n16) * Nn;  // A-fragment row base (per lane)
    const int bcol =  tj * 16 + n16;        // B-fragment / store column

    for (int s = 0; s < CHUNK; ++s) {
        const int  buf      = s & 1;
        const bool havenext = (s + 1 < CHUNK);

        // ---- issue prefetch of M_{s+1} (overlaps the WMMA chain below) ----
        float4 pf0, pf1;
        if (havenext) {
            const float4* src =
                (const float4*)(token_mats + (size_t)ids[s + 1] * (Nn * Nn));
            pf0 = src[tid];
            pf1 = src[tid + 512];
        }

        __syncthreads();   // Mb[buf] committed; previous-step P writes visible

        // ---- C_tile = sum_k P[ti,k] * M[k,tj], K=64 as 16 x (16x16x4) ----
        const float* Mc = Mb[buf];
        v8f acc0 = {};     // even k-steps
        v8f acc1 = {};     // odd  k-steps (breaks WMMA->WMMA RAW chain)
        #pragma unroll
        for (int kt = 0; kt < 16; kt += 2) {
            {
                const int ka = kt * 4 + 2 * half;           // A: VGPR r -> k=2*half+r
                v2f a; a.x = P[arow + ka];       a.y = P[arow + ka + 1];
                v2f b; b.x = Mc[ka * Nn + bcol]; b.y = Mc[(ka + 1) * Nn + bcol];
                acc0 = __builtin_amdgcn_wmma_f32_16x16x4_f32(
                    false, a, false, b, (short)0, acc0, false, false);
            }
            {
                const int ka = (kt + 1) * 4 + 2 * half;
                v2f a; a.x = P[arow + ka];       a.y = P[arow + ka + 1];
                v2f b; b.x = Mc[ka * Nn + bcol]; b.y = Mc[(ka + 1) * Nn + bcol];
                acc1 = __builtin_amdgcn_wmma_f32_16x16x4_f32(
                    false, a, false, b, (short)0, acc1, false, false);
            }
        }
        __syncthreads();   // all reads of P / Mb[buf] complete

        // ---- write D tile back into P (C/D layout: VGPR r -> row r+8*half) ----
        #pragma unroll
        for (int r = 0; r < 8; ++r)
            P[(ti * 16 + r + 8 * half) * Nn + bcol] = acc0[r] + acc1[r];

        // ---- commit prefetched matrix to the alternate buffer ----
        if (havenext) {
            float4* dst = (float4*)Mb[buf ^ 1];
            dst[tid]       = pf0;
            dst[tid + 512] = pf1;
        }
    }
    __syncthreads();

    // ---- emit block-local product ----
    const float4* ps = (const float4*)P;
    float4* pd = (float4*)(partials + (size_t)blockIdx.x * (Nn * Nn));
    pd[tid]       = ps[tid];
    pd[tid + 512] = ps[tid + 512];
}

// ---------------------------------------------------------------------------
// Phase 2: single block folds the NBLK partial products into the query vector
// right-to-left (v = Q_0 (Q_1 (... (Q_{G-1} q)))), then computes logits.
// ---------------------------------------------------------------------------
__global__ __launch_bounds__(256) void chain_finish_kernel(
    const float* __restrict__ partials,     // [NBLK, 64, 64]
    const float* __restrict__ decode_vecs,  // [12, 64]
    const float* __restrict__ query,        // [64]
    float*       __restrict__ out)          // [12]
{
    __shared__ float Qs[Nn * Nn];
    __shared__ float v [Nn];

    const int tid = threadIdx.x;
    if (tid < Nn) v[tid] = query[tid];
    __syncthreads();

    for (int g = NBLK - 1; g >= 0; --g) {
        const float4* src = (const float4*)(partials + (size_t)g * (Nn * Nn));
        for (int u = tid; u < (Nn * Nn) / 4; u += 256) ((float4*)Qs)[u] = src[u];
        __syncthreads();

        float s = 0.0f;
        if (tid < Nn) {
            #pragma unroll 8
            for (int j = 0; j < Nn; ++j) s += Qs[tid * Nn + j] * v[j];
        }
        __syncthreads();
        if (tid < Nn) v[tid] = s;
        __syncthreads();
    }

    if (tid < VOCAB) {
        float s = 0.0f;
        #pragma unroll 8
        for (int j = 0; j < Nn; ++j) s += decode_vecs[tid * Nn + j] * v[j];
        out[tid] = s;
    }
}

// ---------------------------------------------------------------------------
extern "C" void kernel_launch(void* const* d_in, const int* in_sizes, int n_in,
                              void* d_out, int out_size, void* d_ws, size_t ws_size,
                              hipStream_t stream)
{
    (void)in_sizes; (void)n_in; (void)out_size; (void)ws_size;

    const float* token_mats  = (const float*)d_in[0];  // [12,64,64] f32
    const float* decode_vecs = (const float*)d_in[1];  // [12,64]    f32
    const float* query       = (const float*)d_in[2];  // [64]       f32
    const int*   token_ids   = (const int*)  d_in[3];  // [16384]    i32

    float* partials = (float*)d_ws;                    // NBLK*64*64*4 = 2 MB

    chain_partial_kernel<<<NBLK, 512, 0, stream>>>(token_mats, token_ids, partials);
    chain_finish_kernel<<<1, 256, 0, stream>>>(partials, decode_vecs, query,
                                               (float*)d_out);
}